// GRT_51539608009
// MI455X (gfx1250) — compile-verified
//
#include <hip/hip_runtime.h>
#include <hip/hip_bf16.h>

// MVN compound-symmetric log-prob: heavy part is sum(x) and dot(x,x) over 2^25 f32.
// Memory-bound: 128 MiB @ 23.3 TB/s ~ 5.8us. Single pass, b128 loads, WMMA f32
// 16x16x4 (lossless f32) used as the in-register accumulation tree.

typedef float v2f __attribute__((ext_vector_type(2)));
typedef float v8f __attribute__((ext_vector_type(8)));

#define RED_BLOCKS  1024
#define RED_THREADS 256
#define LOG_2PI_F   1.8378770664093453f

// D[m][n] = sum_k A[m][k]*B[k][n] + C[m][n].  With B == ones(4x16), every f32
// value placed in A is added (replicated into 16 columns) into C.  Sum of all
// 256 C entries == 16 * (true sum of all values fed through A).
static __device__ __forceinline__ v8f wmma_acc64(v8f c, v2f a, v2f ones) {
  return __builtin_amdgcn_wmma_f32_16x16x4_f32(
      /*neg_a=*/false, a, /*neg_b=*/false, ones,
      /*c_mod=*/(short)0, c, /*reuse_a=*/false, /*reuse_b=*/false);
}

__global__ __launch_bounds__(RED_THREADS)
void grt_partial_kernel(const float* __restrict__ x, long long n,
                        float* __restrict__ partial) {
  const long long tid    = (long long)blockIdx.x * blockDim.x + threadIdx.x;
  const long long stride = (long long)gridDim.x * blockDim.x;
  const long long n4     = n >> 2;            // # of float4 elements
  const long long full   = n4 / stride;       // wave-uniform iteration count

  v8f csum = {};
  v8f csq  = {};
  v2f ones; ones[0] = 1.0f; ones[1] = 1.0f;

  const float4* __restrict__ x4 = (const float4*)x;

  // Uniform-trip-count main loop: EXEC stays all-1s (WMMA requirement).
#pragma unroll 2
  for (long long k = 0; k < full; ++k) {
    float4 v = x4[tid + k * stride];
    v2f a0; a0[0] = v.x;       a0[1] = v.y;
    v2f a1; a1[0] = v.z;       a1[1] = v.w;
    v2f q0; q0[0] = v.x * v.x; q0[1] = v.y * v.y;
    v2f q1; q1[0] = v.z * v.z; q1[1] = v.w * v.w;
    csum = wmma_acc64(csum, a0, ones);
    csum = wmma_acc64(csum, a1, ones);
    csq  = wmma_acc64(csq,  q0, ones);
    csq  = wmma_acc64(csq,  q1, ones);
  }

  // Ragged remainder (scalar, may diverge — no WMMA here).
  float tsum = 0.0f, tsq = 0.0f;
  for (long long i = full * stride + tid; i < n4; i += stride) {
    float4 v = x4[i];
    tsum += v.x + v.y + v.z + v.w;
    tsq  += v.x * v.x + v.y * v.y + v.z * v.z + v.w * v.w;
  }
  for (long long i = (n4 << 2) + tid; i < n; i += stride) {
    float v = x[i];
    tsum += v;
    tsq  += v * v;
  }

  // Per-lane collapse of the 8 accumulator VGPRs; undo the 16x replication.
  float ls = 0.0f, lq = 0.0f;
#pragma unroll
  for (int k = 0; k < 8; ++k) { ls += csum[k]; lq += csq[k]; }
  ls = ls * 0.0625f + tsum;
  lq = lq * 0.0625f + tsq;

  // wave32 butterfly reduce.
#pragma unroll
  for (int off = 16; off > 0; off >>= 1) {
    ls += __shfl_xor(ls, off, 32);
    lq += __shfl_xor(lq, off, 32);
  }

  __shared__ float ssum[RED_THREADS / 32];
  __shared__ float ssq[RED_THREADS / 32];
  const int lane = threadIdx.x & 31;
  const int wv   = threadIdx.x >> 5;
  if (lane == 0) { ssum[wv] = ls; ssq[wv] = lq; }
  __syncthreads();
  if (threadIdx.x == 0) {
    float bs = 0.0f, bq = 0.0f;
#pragma unroll
    for (int w = 0; w < RED_THREADS / 32; ++w) { bs += ssum[w]; bq += ssq[w]; }
    partial[2 * blockIdx.x]     = bs;
    partial[2 * blockIdx.x + 1] = bq;
  }
}

__global__ __launch_bounds__(RED_THREADS)
void grt_finalize_kernel(const float* __restrict__ partial, int nparts,
                         const float* __restrict__ log_sw,
                         const float* __restrict__ log_sz,
                         const float* __restrict__ log_sx,
                         float* __restrict__ out, long long n) {
  float ls = 0.0f, lq = 0.0f;
  for (int i = threadIdx.x; i < nparts; i += blockDim.x) {
    ls += partial[2 * i];
    lq += partial[2 * i + 1];
  }
#pragma unroll
  for (int off = 16; off > 0; off >>= 1) {
    ls += __shfl_xor(ls, off, 32);
    lq += __shfl_xor(lq, off, 32);
  }
  __shared__ float ssum[RED_THREADS / 32];
  __shared__ float ssq[RED_THREADS / 32];
  const int lane = threadIdx.x & 31;
  const int wv   = threadIdx.x >> 5;
  if (lane == 0) { ssum[wv] = ls; ssq[wv] = lq; }
  __syncthreads();
  if (threadIdx.x == 0) {
    float s = 0.0f, xx = 0.0f;
#pragma unroll
    for (int w = 0; w < RED_THREADS / 32; ++w) { s += ssum[w]; xx += ssq[w]; }

    const float b  = __expf(2.0f * log_sw[0]);
    const float a  = __expf(2.0f * log_sz[0]) + __expf(2.0f * log_sx[0]);
    const float fn = (float)n;
    const float quad   = xx / a - b * s * s / (a * (a + fn * b));
    const float logdet = (fn - 1.0f) * __logf(a) + __logf(a + fn * b);
    out[0] = -0.5f * (fn * LOG_2PI_F + logdet + quad);
  }
}

extern "C" void kernel_launch(void* const* d_in, const int* in_sizes, int n_in,
                              void* d_out, int out_size, void* d_ws, size_t ws_size,
                              hipStream_t stream) {
  const float* x      = (const float*)d_in[0];
  const float* log_sw = (const float*)d_in[1];
  const float* log_sz = (const float*)d_in[2];
  const float* log_sx = (const float*)d_in[3];
  const long long n   = (long long)in_sizes[0];
  float* partial      = (float*)d_ws;  // RED_BLOCKS * 2 floats = 8 KB
  float* out          = (float*)d_out;

  grt_partial_kernel<<<RED_BLOCKS, RED_THREADS, 0, stream>>>(x, n, partial);
  grt_finalize_kernel<<<1, RED_THREADS, 0, stream>>>(partial, RED_BLOCKS,
                                                     log_sw, log_sz, log_sx,
                                                     out, n);
}